// SetAggregationModel_81853486727248
// MI455X (gfx1250) — compile-verified
//
#include <hip/hip_runtime.h>

// ---------------------------------------------------------------------------
// SetAggregationModel: y_a = x_a@W_a + b_a ; y_b = x_b@W_b + b_b ;
// out = concat(segment_mean(y_a, ptr_a), segment_mean(y_b, ptr_b))  [B, 512]
//
// Fused, memory-bound design (HBM floor ~46us on MI455X: 1.07 GB of x reads
// at 23.3 TB/s; W/ptr are L2-resident):
//   kernel 1: zero segment-sum workspace [4096, 512] f32
//   kernel 2: fused GEMM (V_WMMA_F32_16X16X4_F32, 64x64 register tile per
//             wave for ~21 FLOP/B L1 intensity) + in-LDS segmented reduction
//             + sparse atomicAdd of per-segment partial sums
//   kernel 3: normalize by counts and add bias (exact 0 for empty segments)
// ---------------------------------------------------------------------------

typedef __attribute__((ext_vector_type(2))) float v2f;
typedef __attribute__((ext_vector_type(8))) float v8f;

#define D_IN   128
#define D_OUT  256
#define NSEG   4096          // B (num graphs)
#define MTILE  64            // node rows per block
#define OUTW   (2 * D_OUT)   // 512
#define NTHREADS 128         // 4 waves

// --------------------------- zero workspace --------------------------------
__global__ void zero_ws_kernel(float* __restrict__ p, int n) {
    int i = blockIdx.x * blockDim.x + threadIdx.x;
    if (i < n) p[i] = 0.0f;
}

// ------------------- fused GEMM + segment accumulation ---------------------
// grid.x = N/64 (row tiles), grid.y = 2 (node type a / b), block = 128 (4 waves)
// Block tile: 64 rows x 256 cols. Wave w owns cols [64w, 64w+64), all 64 rows:
// a 4x4 grid of 16x16 WMMA tiles (16 v8f accumulators).
__global__ __launch_bounds__(NTHREADS)
void fused_gemm_segsum_kernel(const float* __restrict__ xa,
                              const float* __restrict__ Wa,
                              const int*   __restrict__ pa,
                              const float* __restrict__ xb,
                              const float* __restrict__ Wb,
                              const int*   __restrict__ pb,
                              float* __restrict__ sums) {
    const int type = blockIdx.y;
    const float* __restrict__ x   = type ? xb : xa;
    const float* __restrict__ W   = type ? Wb : Wa;
    const int*   __restrict__ ptr = type ? pb : pa;
    const int col_base = type * D_OUT;   // offset into [.,512] sums

    const int tid  = threadIdx.x;
    const int wave = tid >> 5;           // 0..3
    const int lane = tid & 31;
    const int half = lane >> 4;          // 0: lanes 0-15, 1: lanes 16-31
    const int mrow = lane & 15;
    const int row_base = blockIdx.x * MTILE;

    __shared__ float tile[MTILE * D_OUT];   // 64 KB
    __shared__ int   seg_sh[MTILE];

    // per-row segment id: searchsorted(ptr, row, 'right') - 1
    if (tid < MTILE) {
        const int i = row_base + tid;
        int lo = 0, hi = NSEG;              // invariant: ptr[lo] <= i < ptr[hi]
        while (hi - lo > 1) {
            const int mid = (lo + hi) >> 1;
            if (ptr[mid] <= i) lo = mid; else hi = mid;
        }
        seg_sh[tid] = lo;
    }

    // ---- GEMM ----
    // A fragment (16x4 f32): lanes 0-15 hold K={0,1}, lanes 16-31 K={2,3}
    const float* arow[4];
#pragma unroll
    for (int rt = 0; rt < 4; ++rt)
        arow[rt] = x + (size_t)(row_base + rt * 16 + mrow) * D_IN + half * 2;

    const int colw = wave * 64 + mrow;      // this lane's column in tiles ct*16
    v8f acc[4][4] = {};

#pragma unroll 2
    for (int k0 = 0; k0 < D_IN; k0 += 4) {
        v2f a[4];
#pragma unroll
        for (int rt = 0; rt < 4; ++rt)
            a[rt] = *(const v2f*)(arow[rt] + k0);

        // B fragment (4x16 f32): lane holds W[kb][col], W[kb+1][col]
        const int kb = (k0 + half * 2) * D_OUT;
        v2f b[4];
#pragma unroll
        for (int ct = 0; ct < 4; ++ct) {
            b[ct].x = W[kb + colw + ct * 16];
            b[ct].y = W[kb + D_OUT + colw + ct * 16];
        }

#pragma unroll
        for (int rt = 0; rt < 4; ++rt) {
#pragma unroll
            for (int ct = 0; ct < 4; ++ct) {
                acc[rt][ct] = __builtin_amdgcn_wmma_f32_16x16x4_f32(
                    false, a[rt], false, b[ct], (short)0, acc[rt][ct],
                    false, false);
            }
        }
    }

    // C/D layout: VGPR j -> row (j + half*8) within tile, col = lane&15
#pragma unroll
    for (int rt = 0; rt < 4; ++rt) {
#pragma unroll
        for (int ct = 0; ct < 4; ++ct) {
#pragma unroll
            for (int j = 0; j < 8; ++j) {
                const int r = rt * 16 + j + half * 8;
                tile[r * D_OUT + colw + ct * 16] = acc[rt][ct][j];
            }
        }
    }
    __syncthreads();

    // ---- segmented reduction: 128 threads sweep 256 columns ---------------
    for (int c = tid; c < D_OUT; c += NTHREADS) {
        float run = 0.0f;
        int cur = seg_sh[0];
        for (int r = 0; r < MTILE; ++r) {
            const int s = seg_sh[r];
            if (s != cur) {
                atomicAdd(&sums[(size_t)cur * OUTW + col_base + c], run);
                run = 0.0f;
                cur = s;
            }
            run += tile[r * D_OUT + c];
        }
        atomicAdd(&sums[(size_t)cur * OUTW + col_base + c], run);
    }
}

// ------------------------- normalize + bias --------------------------------
__global__ void finalize_kernel(const float* __restrict__ sums,
                                const int*   __restrict__ pa,
                                const int*   __restrict__ pb,
                                const float* __restrict__ ba,
                                const float* __restrict__ bb,
                                float* __restrict__ out, int n) {
    int idx = blockIdx.x * blockDim.x + threadIdx.x;
    if (idx >= n) return;
    const int g = idx >> 9;         // / 512
    const int c = idx & (OUTW - 1); // % 512
    const int*   ptr  = (c < D_OUT) ? pa : pb;
    const float* bias = (c < D_OUT) ? ba : bb;
    const int cb  = c & (D_OUT - 1);
    const int cnt = ptr[g + 1] - ptr[g];
    float v = 0.0f;
    if (cnt > 0) v = sums[idx] / (float)cnt + bias[cb];
    out[idx] = v;
}

// ---------------------------------------------------------------------------
extern "C" void kernel_launch(void* const* d_in, const int* in_sizes, int n_in,
                              void* d_out, int out_size, void* d_ws, size_t ws_size,
                              hipStream_t stream) {
    const float* xa = (const float*)d_in[0];
    const float* Wa = (const float*)d_in[1];
    const float* ba = (const float*)d_in[2];
    const float* xb = (const float*)d_in[3];
    const float* Wb = (const float*)d_in[4];
    const float* bb = (const float*)d_in[5];
    const int*   pa = (const int*)d_in[6];
    const int*   pb = (const int*)d_in[7];

    float* sums = (float*)d_ws;           // [4096, 512] f32 = 8.4 MB
    float* out  = (float*)d_out;          // [4096, 512] f32

    const int N = in_sizes[0] / D_IN;     // 524288
    const int total = NSEG * OUTW;        // 2,097,152

    zero_ws_kernel<<<(total + 255) / 256, 256, 0, stream>>>(sums, total);

    dim3 grid(N / MTILE, 2);
    fused_gemm_segsum_kernel<<<grid, NTHREADS, 0, stream>>>(xa, Wa, pa,
                                                            xb, Wb, pb, sums);

    finalize_kernel<<<(total + 255) / 256, 256, 0, stream>>>(sums, pa, pb,
                                                             ba, bb, out, total);
}